// StraightThroughPolicy2D_62345745268840
// MI455X (gfx1250) — compile-verified
//
#include <hip/hip_runtime.h>
#include <stdint.h>

typedef __attribute__((ext_vector_type(2))) float v2f;
typedef __attribute__((ext_vector_type(4))) float v4f;
typedef __attribute__((ext_vector_type(8))) float v8f;

#define THREADS 256
#define CHUNK   4096   // elements per block in pass 1 (256 thr * 4 float4)

// sigmoid(2x) = 1/(1+exp(-2x)); exp(-2x) = exp2(-2*log2(e)*x)
__device__ __forceinline__ float sigmoid2(float x) {
    float e = __builtin_amdgcn_exp2f(x * -2.8853900817779268f);
    return __builtin_amdgcn_rcpf(1.0f + e);
}

// deterministic counter-based uniform in [0,1)
__device__ __forceinline__ float u01(uint32_t x) {
    x ^= x >> 17; x *= 0xed5ad4bbu;
    x ^= x >> 11; x *= 0xac4c1b51u;
    x ^= x >> 15; x *= 0x31848babu;
    x ^= x >> 14;
    return (float)(x >> 8) * (1.0f / 16777216.0f);
}

// -------- Pass 1: per-block partial sums of sigmoid(2x)*valid and of valid --------
__global__ void k_partial(const float* __restrict__ x, const uint8_t* __restrict__ mask,
                          float* __restrict__ pP, float* __restrict__ pV,
                          int N, int blocksPerBatch) {
    const int b = blockIdx.y;
    const int c = blockIdx.x;
    const long long base = (long long)b * N + (long long)c * CHUNK;
    const int t = threadIdx.x;

    float sp = 0.0f, sv = 0.0f;
#pragma unroll
    for (int i = 0; i < CHUNK / (THREADS * 4); ++i) {
        const long long idx = base + (long long)(i * THREADS + t) * 4;
        const float4  v  = *(const float4*)(x + idx);
        const uint32_t m = *(const uint32_t*)(mask + idx);
        const float v0 = (m & 0x000000ffu) ? 0.0f : 1.0f;
        const float v1 = (m & 0x0000ff00u) ? 0.0f : 1.0f;
        const float v2 = (m & 0x00ff0000u) ? 0.0f : 1.0f;
        const float v3 = (m & 0xff000000u) ? 0.0f : 1.0f;
        sp += v0 * sigmoid2(v.x) + v1 * sigmoid2(v.y)
            + v2 * sigmoid2(v.z) + v3 * sigmoid2(v.w);
        sv += v0 + v1 + v2 + v3;
    }

    // wave32 butterfly reduction (ds permute path)
#pragma unroll
    for (int o = 16; o >= 1; o >>= 1) {
        sp += __shfl_xor(sp, o, 32);
        sv += __shfl_xor(sv, o, 32);
    }

    __shared__ float lsP[THREADS / 32];
    __shared__ float lsV[THREADS / 32];
    const int wave = t >> 5, lane = t & 31;
    if (lane == 0) { lsP[wave] = sp; lsV[wave] = sv; }
    __syncthreads();
    if (t == 0) {
        float SP = 0.0f, SV = 0.0f;
#pragma unroll
        for (int w = 0; w < THREADS / 32; ++w) { SP += lsP[w]; SV += lsV[w]; }
        pP[b * blocksPerBatch + c] = SP;
        pV[b * blocksPerBatch + c] = SV;
    }
}

// -------- Pass 2: per-batch reduction of partials via V_WMMA_F32_16X16X4_F32 --------
// One wave32 per batch. B matrix = all ones, so D[m][n] = sum_k A[m][k] + C[m][n];
// total sum = sum over column 0 = (component-sum on lane 0) + (component-sum on lane 16).
// Branchless clamped loads keep EXEC all-ones as WMMA requires.
__global__ void k_scale(const float* __restrict__ pP, const float* __restrict__ pV,
                        const int* __restrict__ budget, float4* __restrict__ scale,
                        int blocksPerBatch) {
    const int b    = blockIdx.x;
    const int lane = threadIdx.x;          // 0..31
    const int bbase = b * blocksPerBatch;

    v8f accP = {};
    v8f accV = {};
    v2f ones; ones.x = 1.0f; ones.y = 1.0f;

    const int iters = (blocksPerBatch + 63) / 64;
    for (int it = 0; it < iters; ++it) {
        const int i0 = it * 64 + lane * 2;
        const int j0 = (i0     < blocksPerBatch) ? i0     : (blocksPerBatch - 1);
        const int j1 = (i0 + 1 < blocksPerBatch) ? i0 + 1 : (blocksPerBatch - 1);
        const float w0 = (i0     < blocksPerBatch) ? 1.0f : 0.0f;
        const float w1 = (i0 + 1 < blocksPerBatch) ? 1.0f : 0.0f;
        v2f aP; aP.x = pP[bbase + j0] * w0; aP.y = pP[bbase + j1] * w1;
        v2f aV; aV.x = pV[bbase + j0] * w0; aV.y = pV[bbase + j1] * w1;
        // (neg_a, A, neg_b, B, c_mod, C, reuse_a, reuse_b)
        accP = __builtin_amdgcn_wmma_f32_16x16x4_f32(false, aP, false, ones,
                                                     (short)0, accP, false, false);
        accV = __builtin_amdgcn_wmma_f32_16x16x4_f32(false, aV, false, ones,
                                                     (short)0, accV, false, false);
    }

    float sP = accP[0] + accP[1] + accP[2] + accP[3]
             + accP[4] + accP[5] + accP[6] + accP[7];
    float sV = accV[0] + accV[1] + accV[2] + accV[3]
             + accV[4] + accV[5] + accV[6] + accV[7];
    const float tP = __shfl(sP, 0, 32) + __shfl(sP, 16, 32);
    const float tV = __shfl(sV, 0, 32) + __shfl(sV, 16, 32);

    if (lane == 0) {
        const float n        = tV;
        const float inv_n    = __builtin_amdgcn_rcpf(n);
        const float sparsity = (float)budget[0] * inv_n;
        const float xbar     = tP * inv_n;
        const float r        = sparsity / xbar;
        const float beta     = (1.0f - sparsity) / (1.0f - xbar);
        scale[b] = make_float4(r, beta, (r <= 1.0f) ? 1.0f : 0.0f, 0.0f);
    }
}

// -------- Pass 3: elementwise rescale + straight-through binarize --------
// 2D grid: blockIdx.y = batch (no per-thread 64-bit division).
// Outputs are written with non-temporal stores (th:NT) so the 210MB of
// write-once data streams past L2, keeping x+mask (131MB) L2-resident
// from pass 1 for the re-read here.
__global__ void k_final(const float* __restrict__ x, const uint8_t* __restrict__ mask,
                        const float4* __restrict__ scale,
                        float* __restrict__ outBin, float* __restrict__ outPm,
                        int N) {
    const int b  = blockIdx.y;
    const int o4 = (blockIdx.x * THREADS + threadIdx.x) * 4;  // offset within batch
    if (o4 >= N) return;
    const long long base = (long long)b * N + o4;

    const float4   v  = *(const float4*)(x + base);
    const uint32_t m  = *(const uint32_t*)(mask + base);
    const float4   sc = scale[b];
    const float r = sc.x, beta = sc.y;
    const bool  le = (sc.z != 0.0f);

    float vx[4] = { v.x, v.y, v.z, v.w };
    v4f pm, bn;
#pragma unroll
    for (int j = 0; j < 4; ++j) {
        const float valid = ((m >> (8 * j)) & 0xffu) ? 0.0f : 1.0f;
        const float p = sigmoid2(vx[j]);
        float s = le ? (r * p) : (1.0f - (1.0f - p) * beta);
        s *= valid;
        const float u = u01((uint32_t)(base + j) * 2654435761u + 0x9e3779b9u);
        pm[j] = s;
        bn[j] = (s > u) ? 1.0f : 0.0f;
    }
    __builtin_nontemporal_store(bn, (v4f*)(outBin + base));
    __builtin_nontemporal_store(pm, (v4f*)(outPm + base));
}

extern "C" void kernel_launch(void* const* d_in, const int* in_sizes, int n_in,
                              void* d_out, int out_size, void* d_ws, size_t ws_size,
                              hipStream_t stream) {
    const float*   x      = (const float*)d_in[0];
    const uint8_t* mask   = (const uint8_t*)d_in[1];
    const int*     budget = (const int*)d_in[2];

    const long long total = (long long)in_sizes[0];   // B*H*W = 26,214,400
    const int B = 64;
    const int N = (int)(total / B);                   // 409,600
    const int blocksPerBatch = N / CHUNK;             // 100

    float*  pP    = (float*)d_ws;
    float*  pV    = pP + (size_t)B * blocksPerBatch;
    float4* scale = (float4*)(pV + (size_t)B * blocksPerBatch);

    dim3 g1(blocksPerBatch, B);
    k_partial<<<g1, THREADS, 0, stream>>>(x, mask, pP, pV, N, blocksPerBatch);
    k_scale<<<B, 32, 0, stream>>>(pP, pV, budget, scale, blocksPerBatch);

    const int blocksX = (N / 4 + THREADS - 1) / THREADS;  // 400
    dim3 g3(blocksX, B);
    float* outBin = (float*)d_out;
    float* outPm  = outBin + total;
    k_final<<<g3, THREADS, 0, stream>>>(x, mask, scale, outBin, outPm, N);
}